// EncodeStateDirectly_42734924595829
// MI455X (gfx1250) — compile-verified
//
#include <hip/hip_runtime.h>
#include <hip/hip_bf16.h>

typedef __attribute__((ext_vector_type(2))) float v2f;
typedef __attribute__((ext_vector_type(8))) float v8f;

#define SEQ   48
#define OBS   64
#define HID   84
#define BATCH 32
#define NROWS (SEQ * OBS)   // 3072 (rows of M, size of G)
#define NCOLS (SEQ * HID)   // 4032 (cols of M)
#define PB    64            // batch rows padded to 64 for WMMA tiling
#define CG_ITERS 300

// LDS tile geometry for the WMMA GEMM: 64 rows x 16 K, row stride 20 floats.
// - 80B row stride keeps every b128 async-DMA destination 16B-aligned
// - 20*r mod 64 is injective for r=0..15 -> conflict-free fragment reads
#define LDSSTRIDE 20
#define TILEFL    (64 * LDSSTRIDE)   // floats per tile buffer

// ---------------------------------------------------------------------------
// Workspace layout (in floats)
// ---------------------------------------------------------------------------
static const size_t OFF_CA  = 0;                                   // 48*64*84
static const size_t OFF_M   = OFF_CA + (size_t)SEQ * OBS * HID;    // 3072*4032
static const size_t OFF_G   = OFF_M  + (size_t)NROWS * NCOLS;      // 3072*3072
static const size_t OFF_X   = OFF_G  + (size_t)NROWS * NROWS;      // 64*3072
static const size_t OFF_R   = OFF_X  + (size_t)PB * NROWS;
static const size_t OFF_P   = OFF_R  + (size_t)PB * NROWS;
static const size_t OFF_W   = OFF_P  + (size_t)PB * NROWS;
static const size_t OFF_RRO = OFF_W  + (size_t)PB * NROWS;         // 32
static const size_t OFF_RRN = OFF_RRO + 32;                        // 32
static const size_t OFF_PAP = OFF_RRN + 32;                        // 32

// ---------------------------------------------------------------------------
// CA[k] = C @ A^k   (single block, sequential chain; tiny: 48 * 64x84x84)
// ---------------------------------------------------------------------------
__global__ void build_ca_kernel(const float* __restrict__ C,
                                const float* __restrict__ A,
                                float* __restrict__ CA) {
  const int tid = threadIdx.x;
  for (int idx = tid; idx < OBS * HID; idx += 256) CA[idx] = C[idx];
  __threadfence();
  __syncthreads();
  for (int k = 1; k < SEQ; ++k) {
    const float* prev = CA + (size_t)(k - 1) * OBS * HID;
    float*       cur  = CA + (size_t)k * OBS * HID;
    for (int idx = tid; idx < OBS * HID; idx += 256) {
      const int r = idx / HID;
      const int c = idx - r * HID;
      float acc = 0.f;
      for (int t = 0; t < HID; ++t) acc += prev[r * HID + t] * A[t * HID + c];
      cur[idx] = acc;
    }
    __threadfence();
    __syncthreads();
  }
}

// ---------------------------------------------------------------------------
// Materialize M (3072 x 4032): block (i,j) = CA[i-j] for i>=j, else 0
// ---------------------------------------------------------------------------
__global__ void build_m_kernel(const float* __restrict__ CA,
                               float* __restrict__ Mm) {
  const int idx = blockIdx.x * 256 + threadIdx.x;
  if (idx >= NROWS * NCOLS) return;
  const int r = idx / NCOLS;
  const int c = idx - r * NCOLS;
  const int i = r >> 6;        // row block (OBS=64)
  const int a = r & 63;
  const int j = c / HID;       // col block
  const int h = c - j * HID;
  float v = 0.f;
  if (i >= j) v = CA[(size_t)(i - j) * OBS * HID + a * HID + h];
  Mm[idx] = v;
}

// ---------------------------------------------------------------------------
// Async DMA of one 64x16 f32 tile pair (A rows, B rows) into LDS.
// 128 threads x 16B x 2 per operand = 2 x 4KB. Tracked by ASYNCcnt.
// ---------------------------------------------------------------------------
__device__ __forceinline__ void async_tile_load(const float* __restrict__ Am,
                                                const float* __restrict__ Bm,
                                                int Kdim, int rowBase, int colBase,
                                                int k0, float* ldsA, float* ldsB,
                                                int tid) {
  const int r0 = tid >> 2;           // 0..31
  const int c0 = (tid & 3) * 4;      // 0,4,8,12 (floats)
  const int r1 = r0 + 32;
  unsigned la0 = (unsigned)(uintptr_t)(ldsA + r0 * LDSSTRIDE + c0);
  unsigned la1 = (unsigned)(uintptr_t)(ldsA + r1 * LDSSTRIDE + c0);
  unsigned lb0 = (unsigned)(uintptr_t)(ldsB + r0 * LDSSTRIDE + c0);
  unsigned lb1 = (unsigned)(uintptr_t)(ldsB + r1 * LDSSTRIDE + c0);
  const float* ga0 = Am + (size_t)(rowBase + r0) * Kdim + k0 + c0;
  const float* ga1 = Am + (size_t)(rowBase + r1) * Kdim + k0 + c0;
  const float* gb0 = Bm + (size_t)(colBase + r0) * Kdim + k0 + c0;
  const float* gb1 = Bm + (size_t)(colBase + r1) * Kdim + k0 + c0;
  asm volatile("global_load_async_to_lds_b128 %0, %1, off\n\t"
               "global_load_async_to_lds_b128 %2, %3, off\n\t"
               "global_load_async_to_lds_b128 %4, %5, off\n\t"
               "global_load_async_to_lds_b128 %6, %7, off"
               :
               : "v"(la0), "v"(ga0), "v"(la1), "v"(ga1),
                 "v"(lb0), "v"(gb0), "v"(lb1), "v"(gb1)
               : "memory");
}

__device__ __forceinline__ void wait_async_zero() {
  asm volatile("s_wait_asynccnt 0x0" ::: "memory");
}

// ---------------------------------------------------------------------------
// WMMA f32 GEMM:  Cm (Mdim x Ndim) = Am (Mdim x Kdim) @ Bm (Ndim x Kdim)^T
// All dims multiples of 64. Block = 128 threads (4 waves), 64x64 tile,
// each wave owns a 32x32 quadrant = 2x2 grid of V_WMMA_F32_16X16X4_F32.
// Global->LDS staging uses double-buffered GLOBAL_LOAD_ASYNC_TO_LDS_B128.
// ---------------------------------------------------------------------------
__global__ __launch_bounds__(128)
void gemm_abt_kernel(float* __restrict__ Cm,
                     const float* __restrict__ Am,
                     const float* __restrict__ Bm,
                     int Mdim, int Ndim, int Kdim) {
  __shared__ float lds[4 * TILEFL];  // buf0:{A,B} buf1:{A,B}

  const int tid     = threadIdx.x;
  const int wave    = tid >> 5;
  const int lane    = tid & 31;
  const int waveRow = (wave >> 1) * 32;
  const int waveCol = (wave & 1) * 32;
  const int rowBase = blockIdx.y * 64;
  const int colBase = blockIdx.x * 64;
  const int lm      = lane & 15;          // M/N index within a 16-tile
  const int lk      = (lane >> 4) * 2;    // K sub-offset of this half-wave

  v8f acc[2][2];
  for (int mt = 0; mt < 2; ++mt)
    for (int nt = 0; nt < 2; ++nt)
      for (int p = 0; p < 8; ++p) acc[mt][nt][p] = 0.f;

  const int nChunks = Kdim >> 4;   // K-chunks of 16

  // prologue: DMA chunk 0 into buffer 0
  async_tile_load(Am, Bm, Kdim, rowBase, colBase, 0,
                  lds, lds + TILEFL, tid);

  for (int ci = 0; ci < nChunks; ++ci) {
    // own DMAs for chunk ci done (chunk ci+1 not yet issued)
    wait_async_zero();
    __syncthreads();   // everyone's portion of buffer ci&1 is in LDS

    if (ci + 1 < nChunks) {
      float* nb = lds + ((ci + 1) & 1) * (2 * TILEFL);
      async_tile_load(Am, Bm, Kdim, rowBase, colBase, (ci + 1) << 4,
                      nb, nb + TILEFL, tid);
    }

    const float* As = lds + (ci & 1) * (2 * TILEFL);
    const float* Bs = As + TILEFL;

    for (int kk = 0; kk < 16; kk += 4) {
      for (int mt = 0; mt < 2; ++mt) {
        v2f a;
        a.x = As[(waveRow + mt * 16 + lm) * LDSSTRIDE + kk + lk];
        a.y = As[(waveRow + mt * 16 + lm) * LDSSTRIDE + kk + lk + 1];
        for (int nt = 0; nt < 2; ++nt) {
          v2f b;
          b.x = Bs[(waveCol + nt * 16 + lm) * LDSSTRIDE + kk + lk];
          b.y = Bs[(waveCol + nt * 16 + lm) * LDSSTRIDE + kk + lk + 1];
          acc[mt][nt] = __builtin_amdgcn_wmma_f32_16x16x4_f32(
              false, a, false, b, (short)0, acc[mt][nt], false, false);
        }
      }
    }
  }

  // C/D layout: VGPR p -> M = p (lanes 0-15) / p+8 (lanes 16-31), N = lane%16
  const int rowHalf = (lane >> 4) * 8;
  for (int mt = 0; mt < 2; ++mt)
    for (int nt = 0; nt < 2; ++nt)
      for (int p = 0; p < 8; ++p) {
        const int r = rowBase + waveRow + mt * 16 + rowHalf + p;
        const int c = colBase + waveCol + nt * 16 + lm;
        Cm[(size_t)r * Ndim + c] = acc[mt][nt][p];
      }
}

// ---------------------------------------------------------------------------
// CG helpers
// ---------------------------------------------------------------------------
__global__ void init_cg_kernel(const float* __restrict__ Y,
                               float* __restrict__ X,
                               float* __restrict__ R,
                               float* __restrict__ P) {
  const int idx = blockIdx.x * 256 + threadIdx.x;
  if (idx >= PB * NROWS) return;
  const int row = idx / NROWS;
  const float v = (row < BATCH) ? Y[idx] : 0.f;   // rows 32..63 zero-padded
  X[idx] = 0.f;
  R[idx] = v;
  P[idx] = v;
}

__global__ void rowdot_kernel(float* __restrict__ out,
                              const float* __restrict__ A,
                              const float* __restrict__ B) {
  __shared__ float red[256];
  const int row = blockIdx.x;
  float s = 0.f;
  for (int c = threadIdx.x; c < NROWS; c += 256)
    s += A[(size_t)row * NROWS + c] * B[(size_t)row * NROWS + c];
  red[threadIdx.x] = s;
  __syncthreads();
  for (int off = 128; off > 0; off >>= 1) {
    if (threadIdx.x < off) red[threadIdx.x] += red[threadIdx.x + off];
    __syncthreads();
  }
  if (threadIdx.x == 0) out[row] = red[0];
}

__global__ void update_xr_kernel(float* __restrict__ X, float* __restrict__ R,
                                 const float* __restrict__ P,
                                 const float* __restrict__ W,
                                 const float* __restrict__ rr,
                                 const float* __restrict__ pAp) {
  const int idx = blockIdx.x * 256 + threadIdx.x;
  if (idx >= BATCH * NROWS) return;
  const int row = idx / NROWS;
  const float d = pAp[row];
  const float alpha = (d != 0.f) ? rr[row] / d : 0.f;
  X[idx] += alpha * P[idx];
  R[idx] -= alpha * W[idx];
}

__global__ void update_p_kernel(float* __restrict__ P,
                                const float* __restrict__ R,
                                const float* __restrict__ rr_new,
                                const float* __restrict__ rr_old) {
  const int idx = blockIdx.x * 256 + threadIdx.x;
  if (idx >= BATCH * NROWS) return;
  const int row = idx / NROWS;
  const float d = rr_old[row];
  const float beta = (d != 0.f) ? rr_new[row] / d : 0.f;
  P[idx] = R[idx] + beta * P[idx];
}

__global__ void copy_rr_kernel(float* __restrict__ rr_old,
                               const float* __restrict__ rr_new) {
  const int t = threadIdx.x;
  if (t < BATCH) rr_old[t] = rr_new[t];
}

// loss = sum_b Y_b . X_b / (BATCH * NCOLS)
__global__ void loss_kernel(const float* __restrict__ Y,
                            const float* __restrict__ X,
                            float* __restrict__ out) {
  __shared__ float red[256];
  float s = 0.f;
  for (int idx = threadIdx.x; idx < BATCH * NROWS; idx += 256)
    s += Y[idx] * X[idx];
  red[threadIdx.x] = s;
  __syncthreads();
  for (int off = 128; off > 0; off >>= 1) {
    if (threadIdx.x < off) red[threadIdx.x] += red[threadIdx.x + off];
    __syncthreads();
  }
  if (threadIdx.x == 0) out[0] = red[0] / (float)(BATCH * NCOLS);
}

// ---------------------------------------------------------------------------
// Launch
// ---------------------------------------------------------------------------
extern "C" void kernel_launch(void* const* d_in, const int* in_sizes, int n_in,
                              void* d_out, int out_size, void* d_ws, size_t ws_size,
                              hipStream_t stream) {
  const float* Y = (const float*)d_in[0];  // (32, 3072)
  const float* A = (const float*)d_in[1];  // (84, 84)
  const float* C = (const float*)d_in[2];  // (64, 84)

  float* ws  = (float*)d_ws;
  float* CA  = ws + OFF_CA;
  float* Mm  = ws + OFF_M;
  float* G   = ws + OFF_G;
  float* X   = ws + OFF_X;
  float* R   = ws + OFF_R;
  float* P   = ws + OFF_P;
  float* W   = ws + OFF_W;
  float* rro = ws + OFF_RRO;
  float* rrn = ws + OFF_RRN;
  float* pAp = ws + OFF_PAP;
  float* out = (float*)d_out;

  // 1) CA chain
  build_ca_kernel<<<1, 256, 0, stream>>>(C, A, CA);

  // 2) Materialize M
  {
    const int n = NROWS * NCOLS;
    build_m_kernel<<<(n + 255) / 256, 256, 0, stream>>>(CA, Mm);
  }

  // 3) G = M @ M^T  (WMMA f32, async-DMA double-buffered LDS)
  {
    dim3 grid(NROWS / 64, NROWS / 64);
    gemm_abt_kernel<<<grid, 128, 0, stream>>>(G, Mm, Mm, NROWS, NROWS, NCOLS);
  }

  // 4) CG init:  X=0, R=P=Y (padded to 64 rows), rr_old = R.R
  init_cg_kernel<<<(PB * NROWS + 255) / 256, 256, 0, stream>>>(Y, X, R, P);
  rowdot_kernel<<<BATCH, 256, 0, stream>>>(rro, R, R);

  // 5) CG iterations (fixed count => deterministic, graph-capture safe)
  const int nElem  = BATCH * NROWS;
  const int nBlk   = (nElem + 255) / 256;
  for (int it = 0; it < CG_ITERS; ++it) {
    // W = P @ G  (G symmetric => equals P @ G^T, our A*B^T GEMM)
    dim3 grid(NROWS / 64, PB / 64);
    gemm_abt_kernel<<<grid, 128, 0, stream>>>(W, P, G, PB, NROWS, NROWS);
    rowdot_kernel<<<BATCH, 256, 0, stream>>>(pAp, P, W);
    update_xr_kernel<<<nBlk, 256, 0, stream>>>(X, R, P, W, rro, pAp);
    rowdot_kernel<<<BATCH, 256, 0, stream>>>(rrn, R, R);
    update_p_kernel<<<nBlk, 256, 0, stream>>>(P, R, rrn, rro);
    copy_rr_kernel<<<1, 32, 0, stream>>>(rro, rrn);
  }

  // 6) loss = sum(Y .* X) / (B * S * HID)
  loss_kernel<<<1, 256, 0, stream>>>(Y, X, out);
}